// DistributedRealSHT_63848983822761
// MI455X (gfx1250) — compile-verified
//
#include <hip/hip_runtime.h>

typedef __attribute__((ext_vector_type(2))) float v2f;
typedef __attribute__((ext_vector_type(8))) float v8f;

#define NLAT  360
#define NLON  720
#define LMAX  360
#define MMAX  361
#define NROWS 11520              // 2*16*360 rows for stage-1 GEMM
#define NCOL2 722                // 2*MMAX interleaved re/im columns
#define NCOL2_PAD 736            // padded to 46 col tiles (no load guards)
#define COLT  46
#define ROWT  720                // 11520/16
#define LTILES 23                // ceil(360/16)
#define TWO_PI 6.28318530717958647692f
#define FT_FLOATS ((size_t)NCOL2_PAD * NLON)

#define A_STRIDE 724             // 720 + pad: 16B-aligned rows, conflict-free ds
#define B_STRIDE 364             // 360 + pad: 16B-aligned rows, conflict-free ds

// ---------------------------------------------------------------------------
// CDNA5 async VMEM->LDS copy (ASYNCcnt-tracked). The LDS destination address
// is the low 32 bits of the flat shared pointer (LDS offset).
// ---------------------------------------------------------------------------
__device__ __forceinline__ void async_b128(void* lds_ptr, const void* gptr) {
    uint32_t l = (uint32_t)(uintptr_t)lds_ptr;
    asm volatile("global_load_async_to_lds_b128 %0, %1, off"
                 :: "v"(l), "v"(gptr) : "memory");
}
__device__ __forceinline__ void wait_async0() {
    asm volatile("s_wait_asynccnt 0" ::: "memory");
}

// ---------------------------------------------------------------------------
// ws layout (floats):  Ft[NCOL2_PAD][NLON]  then  Gt[MMAX][64][NLAT]
// ---------------------------------------------------------------------------
__global__ __launch_bounds__(256)
void fill_dft_kernel(float* __restrict__ Ft) {
    int idx = blockIdx.x * 256 + threadIdx.x;
    if (idx >= (int)FT_FLOATS) return;
    int c2 = idx / NLON;                       // interleaved column: 2m + r
    int n  = idx % NLON;
    float v = 0.0f;
    if (c2 < NCOL2) {
        int m  = c2 >> 1;
        int ph = (m * n) % NLON;               // exact phase reduction
        float ang = (TWO_PI / (float)NLON) * (float)ph;
        float s   = TWO_PI / (float)NLON;      // 2*pi * forward-norm scale
        float sv, cv;
        sincosf(ang, &sv, &cv);               // precise path
        v = (c2 & 1) ? (-s * sv) : (s * cv);
    }
    Ft[idx] = v;
}

// ---------------------------------------------------------------------------
// Stage 1: G = X * F.  One block per 16-row A tile; tile async-staged to LDS
// once, 8 waves sweep all 46 col tiles (A from LDS, B from global/L2).
// ---------------------------------------------------------------------------
__global__ __launch_bounds__(256)
void stage1_dft_wmma(const float* __restrict__ X,
                     const float* __restrict__ Ft,
                     float* __restrict__ Gt) {
    __shared__ float Atile[16 * A_STRIDE];     // ~46.3 KB
    const int row0 = blockIdx.x * 16;

    // Async-stage 16 x 720 floats (rows padded to A_STRIDE in LDS).
    for (int t = threadIdx.x; t < 16 * (NLON / 4); t += 256) {
        int rrow = t / (NLON / 4);
        int ch   = t % (NLON / 4);
        async_b128(&Atile[rrow * A_STRIDE + ch * 4],
                   X + (size_t)(row0 + rrow) * NLON + (size_t)ch * 4);
    }
    wait_async0();
    __syncthreads();

    const int lane = threadIdx.x & 31;
    const int wave = threadIdx.x >> 5;
    const int mr   = lane & 15;                // A: M index / B: N index
    const int kh   = (lane >> 4) * 2;          // K sub-offset {0,2}
    const int aoff = mr * A_STRIDE;

    for (int colT = wave; colT < COLT; colT += 8) {   // wave-uniform loop
        const int col0 = colT * 16;
        const float* Brow = Ft + (size_t)(col0 + mr) * NLON; // B[k][N]=Ft[N][k]

        v8f c0 = {0.f,0.f,0.f,0.f,0.f,0.f,0.f,0.f};
        v8f c1 = {0.f,0.f,0.f,0.f,0.f,0.f,0.f,0.f};
        for (int kk = 0; kk < NLON; kk += 8) {
            v2f a0 = *(const v2f*)&Atile[aoff + kk + kh];
            v2f a1 = *(const v2f*)&Atile[aoff + kk + 4 + kh];
            v2f b0 = *(const v2f*)(Brow + kk + kh);
            v2f b1 = *(const v2f*)(Brow + kk + 4 + kh);
            c0 = __builtin_amdgcn_wmma_f32_16x16x4_f32(false, a0, false, b0,
                                                       (short)0, c0, false, false);
            c1 = __builtin_amdgcn_wmma_f32_16x16x4_f32(false, a1, false, b1,
                                                       (short)0, c1, false, false);
        }
        v8f c = c0 + c1;

        // Gt[m][j][k], j = bc*2 + r ; per-lane stores run contiguously in k.
        const int c2 = col0 + mr;              // D: N index
        if (c2 < NCOL2) {
            const int m = c2 >> 1;
            const int r = c2 & 1;
            const int mb = (lane >> 4) * 8;
#pragma unroll
            for (int i = 0; i < 8; ++i) {
                int gr = row0 + mb + i;        // global row = (bc, k)
                int bc = gr / NLAT;
                int k  = gr % NLAT;
                Gt[((size_t)m * 64 + bc * 2 + r) * NLAT + k] = c[i];
            }
        }
    }
}

// ---------------------------------------------------------------------------
// Stage 2: per-m GEMM C_m[l][j] = sum_k W[m][l][k] * Gt[m][j][k].
// One block per (m, j-half): async-stage 32 Gt rows to LDS (reused by all 23
// l tiles); weights are single-use and streamed from global.
// ---------------------------------------------------------------------------
__global__ __launch_bounds__(256)
void stage2_leg_wmma(const float* __restrict__ W,
                     const float* __restrict__ Gt,
                     float* __restrict__ Out) {
    __shared__ float Btile[32 * B_STRIDE];     // ~46.6 KB
    const int m     = blockIdx.x >> 1;
    const int jbase = (blockIdx.x & 1) * 32;

    for (int t = threadIdx.x; t < 32 * (NLAT / 4); t += 256) {
        int rrow = t / (NLAT / 4);
        int ch   = t % (NLAT / 4);
        async_b128(&Btile[rrow * B_STRIDE + ch * 4],
                   Gt + ((size_t)m * 64 + jbase + rrow) * NLAT + (size_t)ch * 4);
    }
    wait_async0();
    __syncthreads();

    const int lane = threadIdx.x & 31;
    const int wave = threadIdx.x >> 5;
    const int mr   = lane & 15;
    const int kh   = (lane >> 4) * 2;

    for (int tile = wave; tile < LTILES * 2; tile += 8) {  // wave-uniform
        const int lT    = tile >> 1;
        const int j0loc = (tile & 1) * 16;
        const int l0    = lT * 16;

        const int  lrow   = l0 + mr;
        const bool avalid = (lrow < LMAX);
        const int  lcl    = avalid ? lrow : (LMAX - 1);
        const float* Arow = W + ((size_t)m * LMAX + lcl) * NLAT;  // W[m][l][k]
        const int  boff   = (j0loc + mr) * B_STRIDE;

        v8f c0 = {0.f,0.f,0.f,0.f,0.f,0.f,0.f,0.f};
        v8f c1 = {0.f,0.f,0.f,0.f,0.f,0.f,0.f,0.f};
        for (int kk = 0; kk < NLAT; kk += 8) {
            v2f a0 = *(const v2f*)(Arow + kk + kh);
            v2f a1 = *(const v2f*)(Arow + kk + 4 + kh);
            v2f b0 = *(const v2f*)&Btile[boff + kk + kh];
            v2f b1 = *(const v2f*)&Btile[boff + kk + 4 + kh];
            a0.x = avalid ? a0.x : 0.f;  a0.y = avalid ? a0.y : 0.f; // cndmask,
            a1.x = avalid ? a1.x : 0.f;  a1.y = avalid ? a1.y : 0.f; // EXEC intact
            c0 = __builtin_amdgcn_wmma_f32_16x16x4_f32(false, a0, false, b0,
                                                       (short)0, c0, false, false);
            c1 = __builtin_amdgcn_wmma_f32_16x16x4_f32(false, a1, false, b1,
                                                       (short)0, c1, false, false);
        }
        v8f c = c0 + c1;

        // Out is complex64 [b][c][l][m] -> interleaved floats; lane pairs
        // (re,im) coalesce to 8B granules.
        const int jcol = jbase + j0loc + mr;   // D: N index
        const int bc   = jcol >> 1;
        const int r    = jcol & 1;
        const int mb   = (lane >> 4) * 8;
#pragma unroll
        for (int i = 0; i < 8; ++i) {
            int l = l0 + mb + i;
            if (l < LMAX) {
                size_t oidx = ((size_t)bc * LMAX + l) * MMAX + m;
                Out[2 * oidx + r] = c[i];
            }
        }
    }
}

extern "C" void kernel_launch(void* const* d_in, const int* in_sizes, int n_in,
                              void* d_out, int out_size, void* d_ws, size_t ws_size,
                              hipStream_t stream) {
    const float* X = (const float*)d_in[0];   // (2,16,360,720) f32
    const float* W = (const float*)d_in[1];   // (361,360,360)  f32
    float* Out = (float*)d_out;               // complex64 -> interleaved f32
    float* Ft  = (float*)d_ws;
    float* Gt  = Ft + FT_FLOATS;

    (void)in_sizes; (void)n_in; (void)out_size; (void)ws_size;

    {   // build DFT basis (2.1 MB, L2-resident afterwards)
        int n = (int)FT_FLOATS;
        fill_dft_kernel<<<(n + 255) / 256, 256, 0, stream>>>(Ft);
    }
    {   // stage 1: one block per 16-row tile
        stage1_dft_wmma<<<ROWT, 256, 0, stream>>>(X, Ft, Gt);
    }
    {   // stage 2: one block per (m, j-half)
        stage2_leg_wmma<<<MMAX * 2, 256, 0, stream>>>(W, Gt, Out);
    }
}